// HeatKernel_45664092291172
// MI455X (gfx1250) — compile-verified
//
#include <hip/hip_runtime.h>
#include <math.h>

typedef __attribute__((ext_vector_type(2))) float v2f;
typedef __attribute__((ext_vector_type(8))) float v8f;

#define N_GENE 60000
#define N_DRUG 40000
#define N_TOT  (N_GENE + N_DRUG)
#define DIM    64
#define N_EDGE 1600000
#define BATCH  4096
#define DECAYF 1.0e-4f

// ---------------------------------------------------------------------------
// agg = concat(gene_embed, drug_embed)   (vectorized b128 copy)
// ---------------------------------------------------------------------------
__global__ void k_concat(const float4* __restrict__ g, const float4* __restrict__ d,
                         float4* __restrict__ agg) {
    int i = blockIdx.x * blockDim.x + threadIdx.x;
    const int n = N_TOT * DIM / 4;
    const int ng = N_GENE * DIM / 4;
    if (i >= n) return;
    agg[i] = (i < ng) ? g[i] : d[i - ng];
}

// ---------------------------------------------------------------------------
// Hop-0: acc[row] = exp(-t)*embed[node]; also per-row sum of squares for reg.
// One 64-thread block per sampled row (user | pos | neg).
// ---------------------------------------------------------------------------
__global__ void k_init_acc(const float* __restrict__ ge, const float* __restrict__ de,
                           const float* __restrict__ gt, const float* __restrict__ dt,
                           const int* __restrict__ user, const int* __restrict__ pos,
                           const int* __restrict__ neg,
                           float* __restrict__ g_acc, float* __restrict__ p_acc,
                           float* __restrict__ n_acc, float* __restrict__ reg_part) {
    int row = blockIdx.x;          // 0 .. 3*BATCH-1
    int d   = threadIdx.x;         // 0..63
    const float* emb; float t; float* acc;
    if (row < BATCH) {
        int u = user[row]; t = gt[u];
        emb = ge + (size_t)u * DIM; acc = g_acc + (size_t)row * DIM;
    } else if (row < 2 * BATCH) {
        int b = row - BATCH; int p = pos[b]; t = dt[p];
        emb = de + (size_t)p * DIM; acc = p_acc + (size_t)b * DIM;
    } else {
        int b = row - 2 * BATCH; int nn = neg[b]; t = dt[nn];
        emb = de + (size_t)nn * DIM; acc = n_acc + (size_t)b * DIM;
    }
    float v = expf(-t) * emb[d];
    acc[d] = v;
    __shared__ float s[64];
    s[d] = v * v;
    __syncthreads();
    for (int off = 32; off; off >>= 1) {
        if (d < off) s[d] += s[d + off];
        __syncthreads();
    }
    if (d == 0) reg_part[row] = s[0];
}

// ---------------------------------------------------------------------------
// SpMM scatter: side[row[e]] += val[e] * agg[col[e]]   (64 f32 atomics / edge)
// 64 consecutive threads handle one edge's 64 dims (coalesced 256B gather).
// ---------------------------------------------------------------------------
__global__ void k_spmm(const float* __restrict__ agg, float* __restrict__ side,
                       const float* __restrict__ val, const int* __restrict__ arow,
                       const int* __restrict__ acol) {
    long tid = (long)blockIdx.x * blockDim.x + threadIdx.x;
    long e = tid >> 6;
    int  d = (int)(tid & 63);
    if (e >= N_EDGE) return;
    if (d == 0) {   // stream the edge lists ahead -> global_prefetch_b8
        __builtin_prefetch(acol + e + 2048, 0, 1);
        __builtin_prefetch(arow + e + 2048, 0, 1);
        __builtin_prefetch(val  + e + 2048, 0, 1);
    }
    int r = arow[e], c = acol[e];
    float v = val[e];
    atomicAdd(side + (long)r * DIM + d, v * agg[(long)c * DIM + d]);
}

// ---------------------------------------------------------------------------
// After hop k: acc[row] += w_k(t[node]) * side[node]
// ---------------------------------------------------------------------------
__global__ void k_hop_acc(const float* __restrict__ side,
                          const float* __restrict__ gt, const float* __restrict__ dt,
                          const int* __restrict__ user, const int* __restrict__ pos,
                          const int* __restrict__ neg,
                          float* __restrict__ g_acc, float* __restrict__ p_acc,
                          float* __restrict__ n_acc, int k, float invfact) {
    int row = blockIdx.x;
    int d   = threadIdx.x;
    float t; long node; float* acc;
    if (row < BATCH) {
        int u = user[row]; t = gt[u]; node = u;
        acc = g_acc + (size_t)row * DIM;
    } else if (row < 2 * BATCH) {
        int b = row - BATCH; int p = pos[b]; t = dt[p]; node = (long)N_GENE + p;
        acc = p_acc + (size_t)b * DIM;
    } else {
        int b = row - 2 * BATCH; int nn = neg[b]; t = dt[nn]; node = (long)N_GENE + nn;
        acc = n_acc + (size_t)b * DIM;
    }
    float pw = 1.0f;
    for (int i = 0; i < k; ++i) pw *= t;
    float w = expf(-t) * pw * invfact;
    acc[d] += w * side[node * DIM + d];
}

// ---------------------------------------------------------------------------
// Batched 64-dim dot products via V_WMMA_F32_16X16X4_F32.
// One wave32 per 16 batch rows; A = g rows (16x64), B = pos/neg rows as KxN.
// A lane L: M = L%16, K pair = {koff, koff+1}, koff = (L>=16)*2.
// B lane L: N = L%16, same K pair (B[k][n] = other[b0+n][k]).
// Diagonal of the 16x16 product = the 16 dot products; extracted via LDS
// using the ISA C/D layout (diag m -> VGPR m%8, lane m (m<8) / m+16 (m>=8)).
// ---------------------------------------------------------------------------
__global__ void k_scores(const float* __restrict__ g_acc, const float* __restrict__ p_acc,
                         const float* __restrict__ n_acc, float* __restrict__ terms) {
    int lane = threadIdx.x;               // 0..31
    int b0   = blockIdx.x * 16;
    int rsel = lane & 15;
    int koff = (lane >> 4) << 1;          // 0 or 2
    const float* gr = g_acc + (size_t)(b0 + rsel) * DIM;
    const float* pr = p_acc + (size_t)(b0 + rsel) * DIM;
    const float* nr = n_acc + (size_t)(b0 + rsel) * DIM;

    v8f cp = {};
    v8f cn = {};
#pragma unroll
    for (int k0 = 0; k0 < DIM; k0 += 4) {
        v2f a, bp, bn;
        a.x  = gr[k0 + koff];  a.y  = gr[k0 + koff + 1];
        bp.x = pr[k0 + koff];  bp.y = pr[k0 + koff + 1];
        bn.x = nr[k0 + koff];  bn.y = nr[k0 + koff + 1];
        cp = __builtin_amdgcn_wmma_f32_16x16x4_f32(false, a, false, bp, (short)0, cp, false, false);
        cn = __builtin_amdgcn_wmma_f32_16x16x4_f32(false, a, false, bn, (short)0, cn, false, false);
    }

    __shared__ float sp[32 * 8];
    __shared__ float sn[32 * 8];
#pragma unroll
    for (int i = 0; i < 8; ++i) {
        sp[lane * 8 + i] = cp[i];
        sn[lane * 8 + i] = cn[i];
    }
    __syncthreads();

    if (lane < 16) {
        int m = lane;
        int srcLane = (m < 8) ? m : (m + 16);
        int vg      = (m < 8) ? m : (m - 8);
        // accumulators hold 4x the hop-mean; /4 per operand => *0.0625 on product
        float ps = sp[srcLane * 8 + vg] * 0.0625f;
        float ns = sn[srcLane * 8 + vg] * 0.0625f;
        float x = ns - ps;                               // K==1
        float term = (x > 0.0f) ? (x + log1pf(expf(-x))) : log1pf(expf(x));
        terms[b0 + m] = term;
    }
}

// ---------------------------------------------------------------------------
// Deterministic final reduction: mf = mean(terms); emb = DECAY*(reg/2)/B.
// ---------------------------------------------------------------------------
__global__ void k_final(const float* __restrict__ terms, const float* __restrict__ reg_part,
                        float* __restrict__ out) {
    __shared__ float s1[256];
    __shared__ float s2[256];
    int t = threadIdx.x;
    float a = 0.0f, b = 0.0f;
    for (int i = t; i < BATCH; i += 256)     a += terms[i];
    for (int i = t; i < 3 * BATCH; i += 256) b += reg_part[i];
    s1[t] = a; s2[t] = b;
    __syncthreads();
    for (int off = 128; off; off >>= 1) {
        if (t < off) { s1[t] += s1[t + off]; s2[t] += s2[t + off]; }
        __syncthreads();
    }
    if (t == 0) {
        float mf  = s1[0] / (float)BATCH;
        float emb = DECAYF * (s2[0] * 0.5f) / (float)BATCH;
        out[0] = mf + emb;
        out[1] = mf;
        out[2] = emb;
    }
}

// ---------------------------------------------------------------------------
extern "C" void kernel_launch(void* const* d_in, const int* in_sizes, int n_in,
                              void* d_out, int out_size, void* d_ws, size_t ws_size,
                              hipStream_t stream) {
    const float* ge   = (const float*)d_in[0];
    const float* de   = (const float*)d_in[1];
    const float* gt   = (const float*)d_in[2];
    const float* dt   = (const float*)d_in[3];
    const float* aval = (const float*)d_in[4];
    const int*   arow = (const int*)d_in[5];
    const int*   acol = (const int*)d_in[6];
    const int*   user = (const int*)d_in[7];
    const int*   pos  = (const int*)d_in[8];
    const int*   neg  = (const int*)d_in[9];
    float* out = (float*)d_out;

    float* ws = (float*)d_ws;
    size_t off = 0;
    float* bufA     = ws + off; off += (size_t)N_TOT * DIM;
    float* bufB     = ws + off; off += (size_t)N_TOT * DIM;
    float* g_acc    = ws + off; off += (size_t)BATCH * DIM;
    float* p_acc    = ws + off; off += (size_t)BATCH * DIM;
    float* n_acc    = ws + off; off += (size_t)BATCH * DIM;
    float* reg_part = ws + off; off += (size_t)BATCH * 3;
    float* terms    = ws + off; off += (size_t)BATCH;

    {   // agg0 = concat(gene, drug)
        int n = N_TOT * DIM / 4;
        k_concat<<<(n + 255) / 256, 256, 0, stream>>>(
            (const float4*)ge, (const float4*)de, (float4*)bufA);
    }
    // hop-0 weighted gather + regularizer partials (fully overwrites acc/reg)
    k_init_acc<<<3 * BATCH, 64, 0, stream>>>(ge, de, gt, dt, user, pos, neg,
                                             g_acc, p_acc, n_acc, reg_part);

    float* agg  = bufA;
    float* side = bufB;
    const float invfact[4] = {1.0f, 1.0f, 0.5f, 1.0f / 6.0f};
    for (int k = 1; k <= 3; ++k) {
        hipMemsetAsync(side, 0, (size_t)N_TOT * DIM * sizeof(float), stream);
        long tot = (long)N_EDGE * DIM;
        k_spmm<<<(int)((tot + 255) / 256), 256, 0, stream>>>(agg, side, aval, arow, acol);
        k_hop_acc<<<3 * BATCH, 64, 0, stream>>>(side, gt, dt, user, pos, neg,
                                                g_acc, p_acc, n_acc, k, invfact[k]);
        float* tmp = agg; agg = side; side = tmp;
    }

    k_scores<<<BATCH / 16, 32, 0, stream>>>(g_acc, p_acc, n_acc, terms);
    k_final<<<1, 256, 0, stream>>>(terms, reg_part, out);
}